// AlwaysLayer_57071525429382
// MI455X (gfx1250) — compile-verified
//
#include <hip/hip_runtime.h>
#include <cstdint>

// ---- problem constants (from the reference) ---------------------------------
#define BIN1   8
#define NRULE  4
#define BIN_T  8
#define KGRP   64        // input_dim / (BIN1*BIN_T)
#define IN_DIM 4096
#define OUT_DIM 2048     // KGRP * BIN1 * NRULE
// ---- tiling -----------------------------------------------------------------
#define ROWS    128      // batch rows per block (1 row per thread)
#define THREADS 128      // 4 waves (wave32)
#define LDS_ROW 65       // 64 data DWORDs + 1 TDM pad DWORD -> conflict-free

typedef uint32_t u32x4 __attribute__((ext_vector_type(4)));
typedef int      i32x4 __attribute__((ext_vector_type(4)));
typedef int      i32x8 __attribute__((ext_vector_type(8)));
typedef float    f32x4 __attribute__((ext_vector_type(4)));   // native vec for NT store

#if __has_builtin(__builtin_amdgcn_tensor_load_to_lds) && \
    __has_builtin(__builtin_amdgcn_s_wait_tensorcnt)
#define USE_TDM 1
#else
#define USE_TDM 0   // host pass / unexpected toolchain: cooperative loads
#endif

__device__ __forceinline__ float fast_log2(float v) {
#if __has_builtin(__builtin_amdgcn_logf)
  return __builtin_amdgcn_logf(v);      // v_log_f32
#else
  return __log2f(v);
#endif
}

__global__ __launch_bounds__(THREADS, 1)
void always_layer_kernel(const float* __restrict__ x,
                         const float* __restrict__ w_start,
                         const float* __restrict__ w_end,
                         float* __restrict__ out) {
  __shared__ float xtile[ROWS * LDS_ROW];              // ~33 KB
  __shared__ float wlds[BIN1 * NRULE * BIN_T];         // 256 floats, this k only

  const int tid = threadIdx.x;
  const int k   = blockIdx.x & (KGRP - 1);
  const int b0  = (blockIdx.x >> 6) * ROWS;

#if USE_TDM
  // -------- one TDM descriptor: 64x128 f32 tile, tensor row stride 4096 elems,
  // -------- LDS padding of 1 DWORD every 64 DWORDs (-> 65-float LDS rows).
  if (tid == 0) {
    const uint64_t gaddr =
        (uint64_t)(uintptr_t)(x + (size_t)b0 * IN_DIM + (size_t)k * (BIN_T * BIN1));
    const uint32_t ldsa = (uint32_t)(uintptr_t)&xtile[0];   // low 32b = LDS offset
    u32x4 g0;
    g0.x = 1u;                                              // count = 1 (valid D#)
    g0.y = ldsa;                                            // lds_addr
    g0.z = (uint32_t)gaddr;                                 // global_addr[31:0]
    g0.w = (uint32_t)((gaddr >> 32) & 0x1FFFFFFu) | (2u << 30); // [56:32] | type=2
    i32x8 g1;
    g1[0] = (2 << 16)      // data_size = 4B
          | (1 << 20)      // pad_enable
          | (5 << 22);     // pad_interval: every 64 DWORDs (pad_amount code 0 = 1 DW)
    g1[1] = (int)(64u << 16);                 // tensor_dim0 = 64   (low16 @ [31:16])
    g1[2] = (int)((uint32_t)ROWS << 16);      // tensor_dim1 = 128  (low16 @ [31:16])
    g1[3] = (int)(64u << 16);                 // tile_dim0   = 64   (@ [31:16])
    g1[4] = ROWS;                             // tile_dim1   = 128  (@ [15:0])
    g1[5] = IN_DIM;                           // tensor_dim0_stride = 4096 elements
    g1[6] = 0;
    g1[7] = 0;
    i32x4 gz = {0, 0, 0, 0};                  // 2D tensor: groups 2/3 unused
#if defined(__clang_major__) && __clang_major__ >= 23
    i32x8 gz8 = {0, 0, 0, 0, 0, 0, 0, 0};
    __builtin_amdgcn_tensor_load_to_lds(g0, g1, gz, gz, gz8, 0);
#else
    __builtin_amdgcn_tensor_load_to_lds(g0, g1, gz, gz, 0);
#endif
  }
#else
  // Fallback: cooperative synchronous fill (16 float4 per row).
  for (int i = tid; i < ROWS * 16; i += THREADS) {
    const int r = i >> 4, c = i & 15;
    f32x4 v = *(const f32x4*)(x + (size_t)(b0 + r) * IN_DIM + k * 64 + c * 4);
    float* dst = &xtile[r * LDS_ROW + c * 4];
    dst[0] = v.x; dst[1] = v.y; dst[2] = v.z; dst[3] = v.w;
  }
#endif

  // -------- membership weights for this k (overlaps the DMA) ----------------
  // wlds[(j*4+n)*8 + t] = sigmoid((t - ws)*(we - t))
  for (int i = tid; i < BIN1 * NRULE * BIN_T; i += THREADS) {
    const int   jn = i >> 3;
    const float t  = (float)(i & 7);
    const float ws = w_start[k * (BIN1 * NRULE) + jn];
    const float we = w_end  [k * (BIN1 * NRULE) + jn];
    const float z  = (t - ws) * (we - t);
    wlds[i] = 1.0f / (1.0f + __expf(-z));     // v_exp_f32 path; 2 per thread
  }

#if USE_TDM
  if (tid == 0) __builtin_amdgcn_s_wait_tensorcnt(0);
#endif
  __syncthreads();

  // -------- per-thread: one batch row, 32 outputs ----------------------------
  // LDS reads: lane L reads xtile[L*65 + idx] -> bank (L+idx)%64: conflict-free.
  // wlds reads are wave-uniform (all lanes share k,j,n,t): broadcast.
  const float* xr   = &xtile[tid * LDS_ROW];
  float*       orow = out + (size_t)(b0 + tid) * OUT_DIM + k * (BIN1 * NRULE);
  const float  LN2  = 0.69314718055994530942f;

#pragma unroll
  for (int j = 0; j < BIN1; ++j) {
    float om[BIN_T];
#pragma unroll
    for (int t = 0; t < BIN_T; ++t) om[t] = 1.0f - xr[t * BIN1 + j];
    f32x4 o;
#pragma unroll
    for (int n = 0; n < NRULE; ++n) {
      const float* wp  = &wlds[(j * NRULE + n) * BIN_T];
      float        acc = 0.0f;
#pragma unroll
      for (int t = 0; t < BIN_T; ++t)
        acc += fast_log2(__builtin_fmaf(-om[t], wp[t], 1.0f)); // log2(1 - om*W)
      // y = -1/(-1 + ln2*acc) = 1/(1 - ln2*acc), acc <= 0
      o[n] = 1.0f / (1.0f - LN2 * acc);
    }
    __builtin_nontemporal_store(o, (f32x4*)(orow + j * 4));   // stream-once output
  }
}

extern "C" void kernel_launch(void* const* d_in, const int* in_sizes, int n_in,
                              void* d_out, int out_size, void* d_ws, size_t ws_size,
                              hipStream_t stream) {
  const float* x  = (const float*)d_in[0];
  const float* ws = (const float*)d_in[1];
  const float* we = (const float*)d_in[2];
  float*       out = (float*)d_out;
  const int B    = in_sizes[0] / IN_DIM;          // 4096
  const int grid = (B / ROWS) * KGRP;             // 32 * 64 = 2048 blocks
  always_layer_kernel<<<grid, THREADS, 0, stream>>>(x, ws, we, out);
}